// SphericalSamplingDecoder_51926154609187
// MI455X (gfx1250) — compile-verified
//
#include <hip/hip_runtime.h>
#include <math.h>

// MI455X / gfx1250, wave32.
//  K1: pack l=4 SH basis (9 comps padded to K=12) of the grid vecs into the
//      V_WMMA_F32_16X16X4_F32 B-operand fragment layout; lane's dwords adjacent
//      => 3 coalesced global_load_b64 (imm offsets 0/256/512) per 16-col tile.
//  K2a/K2c: per-wave 16-row x quarter-column streaming argmax via 3 chained
//      fp32 WMMAs per tile. Scalarized (readfirstlane) loop bounds, pointer
//      strength reduction, WGP-scope prefetch. 4096 waves per pass.
//  K2b/K2d: merge ordered column-split partials (== jnp.argmax first-occurrence).
//  K3: Gram-Schmidt + cross + matrix->quaternion epilogue, plus f0*180/pi.

typedef float v2f __attribute__((ext_vector_type(2)));
typedef float v8f __attribute__((ext_vector_type(8)));

#define NSPLIT 4

__device__ __forceinline__ void prefetch_wgp(const void* p) {
  // scope 0 (WGP): pull into all cache levels on miss (prints without suffix)
  asm volatile("global_prefetch_b8 %0, off" ::"v"(p) : "memory");
}

__device__ __forceinline__ float sh4_comp(int kk, float x, float y, float z) {
  const float c_m4 = 2.5033429417967046f;   // 0.75*sqrt(35/pi)
  const float c_m3 = 1.7701307697799304f;   // 0.75*sqrt(35/(2pi))
  const float c_m2 = 0.9461746957575601f;   // 0.75*sqrt(5/pi)
  const float c_m1 = 0.6690465435572892f;   // 0.75*sqrt(5/(2pi))
  const float c_0  = 0.10578554691520431f;  // (3/16)*sqrt(1/pi)
  const float c_p2 = 0.47308734787878004f;  // (3/8)*sqrt(5/pi)
  const float c_p4 = 0.6258357354491761f;   // (3/16)*sqrt(35/pi)
  float x2 = x * x, y2 = y * y, z2 = z * z;
  switch (kk) {
    case 0: return c_m4 * x * y * (x2 - y2);
    case 1: return c_m3 * y * z * (3.0f * x2 - y2);
    case 2: return c_m2 * x * y * (7.0f * z2 - 1.0f);
    case 3: return c_m1 * y * z * (7.0f * z2 - 3.0f);
    case 4: return c_0 * (35.0f * z2 * z2 - 30.0f * z2 + 3.0f);
    case 5: return c_m1 * x * z * (7.0f * z2 - 3.0f);
    case 6: return c_p2 * (x2 - y2) * (7.0f * z2 - 1.0f);
    case 7: return c_m3 * x * z * (x2 - y2);
    case 8: return c_p4 * (x2 * x2 - 6.0f * x2 * y2 + y2 * y2);
    default: return 0.0f;  // K padding 9..11
  }
}

// ---------------- K1: build packed B-operand panel ----------------
// float2 element index = (t*3 + s)*32 + lane; component v.
// lane<16: K = 4s+v, col N=lane ; lane>=16: K = 4s+v+2, col N=lane-16.
// Value = Y4[g = 16t + N][K]  (mirror of the ISA 32-bit 16x4 A layout).
__global__ void sph_build_ypack(const float* __restrict__ gv,
                                float* __restrict__ ypack, int nE) {
  int e = blockIdx.x * blockDim.x + threadIdx.x;
  if (e >= nE) return;
  int v = e & 1;
  int p = e >> 1;
  int lane = p & 31;
  int sv = p >> 5;           // = t*3 + s
  int s = sv % 3;
  int t = sv / 3;
  int n = lane & 15;
  int kk = 4 * s + v + ((lane >= 16) ? 2 : 0);
  int g = t * 16 + n;
  float x = gv[g * 3 + 0], y = gv[g * 3 + 1], z = gv[g * 3 + 2];
  float nn = fmaxf(sqrtf(x * x + y * y + z * z), 1e-12f);
  x /= nn; y /= nn; z /= nn;
  ypack[e] = sh4_comp(kk, x, y, z);
}

// Butterfly argmax over the 16 lanes sharing a row (xor 8,4,2,1 stays in-half).
// Ties -> smaller grid index (matches jnp.argmax first-occurrence).
__device__ __forceinline__ void reduce_argmax16(float& v, int& i) {
#pragma unroll
  for (int off = 8; off; off >>= 1) {
    float ov = __shfl_xor(v, off, 32);
    int   oi = __shfl_xor(i, off, 32);
    if (ov > v || (ov == v && oi < i)) { v = ov; i = oi; }
  }
}

// Load the A fragment: rows b = m*16 + M of f4, K padded 9->12 with zeros.
// lane<16: M=lane, K=4s+v ; lane>=16: M=lane-16, K=4s+v+2.
__device__ __forceinline__ void load_afrag(const float* __restrict__ f4,
                                           int brow, int hi, v2f a[3]) {
#pragma unroll
  for (int s = 0; s < 3; ++s) {
#pragma unroll
    for (int v = 0; v < 2; ++v) {
      int kk = 4 * s + v + hi * 2;
      a[s][v] = (kk < 9) ? f4[brow * 9 + kk] : 0.0f;
    }
  }
}

// ---------------- K2a: pass-1 partial argmax (column split) ----------------
__global__ __launch_bounds__(256) void sph_pass1(
    const float* __restrict__ f4, const float* __restrict__ ypack,
    float* __restrict__ pv, int* __restrict__ pidx,
    int GT, int B, int nWaves) {
  const int lane = threadIdx.x & 31;
  // wave-uniform -> force into SGPRs so the hot loop is a scalar loop
  const int w = __builtin_amdgcn_readfirstlane(blockIdx.x * 8 + (threadIdx.x >> 5));
  if (w >= nWaves) return;
  const int split = w % NSPLIT;
  const int m = w / NSPLIT;                // 16-row tile
  const int hi = lane >> 4;
  const int M = lane & 15;
  const int t0 = (split * GT) / NSPLIT;
  const int t1 = ((split + 1) * GT) / NSPLIT;
  const int nt = t1 - t0;                  // scalar trip count

  v2f a[3];
  load_afrag(f4, m * 16 + M, hi, a);

  const char* p = (const char*)ypack + (size_t)t0 * 768 + (size_t)lane * 8;
  int g = t0 * 16 + M;

  float bv[8]; int bidx[8];
#pragma unroll
  for (int r = 0; r < 8; ++r) { bv[r] = -__builtin_inff(); bidx[r] = 0; }

  for (int it = 0; it < nt; ++it) {
    prefetch_wgp(p + 8 * 768);          // lands in ws even past ypack end: safe
    v2f b0 = *(const v2f*)(p);
    v2f b1 = *(const v2f*)(p + 256);
    v2f b2 = *(const v2f*)(p + 512);
    v8f c = {0.f, 0.f, 0.f, 0.f, 0.f, 0.f, 0.f, 0.f};
    c = __builtin_amdgcn_wmma_f32_16x16x4_f32(false, a[0], false, b0, (short)0, c, false, false);
    c = __builtin_amdgcn_wmma_f32_16x16x4_f32(false, a[1], false, b1, (short)0, c, false, false);
    c = __builtin_amdgcn_wmma_f32_16x16x4_f32(false, a[2], false, b2, (short)0, c, false, false);
#pragma unroll
    for (int r = 0; r < 8; ++r)
      if (c[r] > bv[r]) { bv[r] = c[r]; bidx[r] = g; }
    g += 16;
    p += 768;
  }

#pragma unroll
  for (int r = 0; r < 8; ++r) {
    reduce_argmax16(bv[r], bidx[r]);
    if (M == 0) {
      int row = m * 16 + r + hi * 8;
      pv[split * B + row] = bv[r];
      pidx[split * B + row] = bidx[r];
    }
  }
}

// ---------------- K2b/K2d: merge column-split partials ----------------
__global__ void sph_merge(const float* __restrict__ pv, const int* __restrict__ pidx,
                          int* __restrict__ outIdx, int B) {
  int b = blockIdx.x * blockDim.x + threadIdx.x;
  if (b >= B) return;
  float best = pv[b];
  int bi = pidx[b];
#pragma unroll
  for (int s = 1; s < NSPLIT; ++s) {   // split ranges ordered => strict > keeps
    float v = pv[s * B + b];           // the first occurrence of the max
    int i = pidx[s * B + b];
    if (v > best) { best = v; bi = i; }
  }
  outIdx[b] = bi;
}

// ---------------- K2c: pass-2 masked partial argmax ----------------
__global__ __launch_bounds__(256) void sph_pass2(
    const float* __restrict__ f4, const float* __restrict__ gv,
    const float* __restrict__ ypack, const int* __restrict__ zi,
    float* __restrict__ pv, int* __restrict__ pidx,
    int GT, int B, int nWaves) {
  const int lane = threadIdx.x & 31;
  const int w = __builtin_amdgcn_readfirstlane(blockIdx.x * 8 + (threadIdx.x >> 5));
  if (w >= nWaves) return;
  const int split = w % NSPLIT;
  const int m = w / NSPLIT;
  const int hi = lane >> 4;
  const int M = lane & 15;
  const int t0 = (split * GT) / NSPLIT;
  const int t1 = ((split + 1) * GT) / NSPLIT;
  const int nt = t1 - t0;

  v2f a[3];
  load_afrag(f4, m * 16 + M, hi, a);

  float zax[8], zay[8], zaz[8];
#pragma unroll
  for (int r = 0; r < 8; ++r) {
    int zg = zi[m * 16 + r + hi * 8];
    zax[r] = gv[zg * 3 + 0];
    zay[r] = gv[zg * 3 + 1];
    zaz[r] = gv[zg * 3 + 2];
  }

  const char* p = (const char*)ypack + (size_t)t0 * 768 + (size_t)lane * 8;
  const float* gvp = gv + (size_t)(t0 * 16 + M) * 3;
  int g = t0 * 16 + M;

  float bv[8]; int bidx[8];
#pragma unroll
  for (int r = 0; r < 8; ++r) { bv[r] = -__builtin_inff(); bidx[r] = 0; }

  for (int it = 0; it < nt; ++it) {
    prefetch_wgp(p + 8 * 768);
    v2f b0 = *(const v2f*)(p);
    v2f b1 = *(const v2f*)(p + 256);
    v2f b2 = *(const v2f*)(p + 512);
    float gx = gvp[0], gy = gvp[1], gz = gvp[2];
    v8f c = {0.f, 0.f, 0.f, 0.f, 0.f, 0.f, 0.f, 0.f};
    c = __builtin_amdgcn_wmma_f32_16x16x4_f32(false, a[0], false, b0, (short)0, c, false, false);
    c = __builtin_amdgcn_wmma_f32_16x16x4_f32(false, a[1], false, b1, (short)0, c, false, false);
    c = __builtin_amdgcn_wmma_f32_16x16x4_f32(false, a[2], false, b2, (short)0, c, false, false);
#pragma unroll
    for (int r = 0; r < 8; ++r) {
      float d = zax[r] * gx + zay[r] * gy + zaz[r] * gz;
      float val = (fabsf(d) < 0.2f) ? c[r] : -__builtin_inff();
      if (val > bv[r]) { bv[r] = val; bidx[r] = g; }
    }
    g += 16;
    p += 768;
    gvp += 48;
  }

#pragma unroll
  for (int r = 0; r < 8; ++r) {
    reduce_argmax16(bv[r], bidx[r]);
    if (M == 0) {
      int row = m * 16 + r + hi * 8;
      pv[split * B + row] = bv[r];
      pidx[split * B + row] = bidx[r];
    }
  }
}

// ---------------- K3: per-row epilogue ----------------
__global__ void sph_finalize(const float* __restrict__ gv, const float* __restrict__ f0,
                             const int* __restrict__ zi, const int* __restrict__ xi,
                             float* __restrict__ out, int B) {
  int b = blockIdx.x * blockDim.x + threadIdx.x;
  if (b >= B) return;
  int zg = zi[b], xg = xi[b];
  float zx = gv[zg * 3 + 0], zy = gv[zg * 3 + 1], zz = gv[zg * 3 + 2];
  float ax = gv[xg * 3 + 0], ay = gv[xg * 3 + 1], az = gv[xg * 3 + 2];
  float zn = fmaxf(sqrtf(zx * zx + zy * zy + zz * zz), 1e-12f);
  zx /= zn; zy /= zn; zz /= zn;
  float d = ax * zx + ay * zy + az * zz;          // project x off z
  ax -= d * zx; ay -= d * zy; az -= d * zz;
  float an = fmaxf(sqrtf(ax * ax + ay * ay + az * az), 1e-12f);
  ax /= an; ay /= an; az /= an;
  float yx = zy * az - zz * ay;                   // y = z cross x
  float yy = zz * ax - zx * az;
  float yz = zx * ay - zy * ax;
  // R columns = (x, y, z)
  float m00 = ax, m01 = yx, m02 = zx;
  float m10 = ay, m11 = yy, m12 = zy;
  float m20 = az, m21 = yz, m22 = zz;
  float t0 = fmaxf(1.f + m00 + m11 + m22, 0.f);
  float t1 = fmaxf(1.f + m00 - m11 - m22, 0.f);
  float t2 = fmaxf(1.f - m00 + m11 - m22, 0.f);
  float t3 = fmaxf(1.f - m00 - m11 + m22, 0.f);
  float q0 = sqrtf(t0), q1 = sqrtf(t1), q2 = sqrtf(t2), q3 = sqrtf(t3);
  int best = 0; float bq = q0;
  if (q1 > bq) { bq = q1; best = 1; }
  if (q2 > bq) { bq = q2; best = 2; }
  if (q3 > bq) { bq = q3; best = 3; }
  float n0, n1, n2, n3;
  if (best == 0)      { n0 = t0;        n1 = m21 - m12; n2 = m02 - m20; n3 = m10 - m01; }
  else if (best == 1) { n0 = m21 - m12; n1 = t1;        n2 = m10 + m01; n3 = m02 + m20; }
  else if (best == 2) { n0 = m02 - m20; n1 = m10 + m01; n2 = t2;        n3 = m12 + m21; }
  else                { n0 = m10 - m01; n1 = m20 + m02; n2 = m21 + m12; n3 = t3; }
  float inv = 1.0f / (2.0f * fmaxf(bq, 0.1f));
  out[b * 4 + 0] = n0 * inv;
  out[b * 4 + 1] = n1 * inv;
  out[b * 4 + 2] = n2 * inv;
  out[b * 4 + 3] = n3 * inv;
  out[(size_t)B * 4 + b] = f0[b] * 57.29577951308232f;  // 180/pi
}

extern "C" void kernel_launch(void* const* d_in, const int* in_sizes, int n_in,
                              void* d_out, int out_size, void* d_ws, size_t ws_size,
                              hipStream_t stream) {
  const float* f0 = (const float*)d_in[0];
  const float* f4 = (const float*)d_in[2];
  const float* gv = (const float*)d_in[4];
  const int B  = in_sizes[0];          // 16384
  const int G  = in_sizes[4] / 3;      // 10000 (divisible by 16)
  const int GT = G / 16;               // 625 column tiles

  // workspace layout
  char* ws = (char*)d_ws;
  float* ypack = (float*)ws;                         ws += (size_t)GT * 192 * sizeof(float);
  int*   zidx  = (int*)ws;                           ws += (size_t)B * sizeof(int);
  int*   xidx  = (int*)ws;                           ws += (size_t)B * sizeof(int);
  float* pv    = (float*)ws;                         ws += (size_t)NSPLIT * B * sizeof(float);
  int*   pidx  = (int*)ws;

  const int nE = GT * 192;
  sph_build_ypack<<<(nE + 255) / 256, 256, 0, stream>>>(gv, ypack, nE);

  const int nWaves = (B / 16) * NSPLIT;      // 4096 waves per pass
  const int nBlocks = (nWaves + 7) / 8;

  sph_pass1<<<nBlocks, 256, 0, stream>>>(f4, ypack, pv, pidx, GT, B, nWaves);
  sph_merge<<<(B + 255) / 256, 256, 0, stream>>>(pv, pidx, zidx, B);
  sph_pass2<<<nBlocks, 256, 0, stream>>>(f4, gv, ypack, zidx, pv, pidx, GT, B, nWaves);
  sph_merge<<<(B + 255) / 256, 256, 0, stream>>>(pv, pidx, xidx, B);
  sph_finalize<<<(B + 255) / 256, 256, 0, stream>>>(gv, f0, zidx, xidx, (float*)d_out, B);
}